// KDeepSeekV3Cache_42906723287457
// MI455X (gfx1250) — compile-verified
//
#include <hip/hip_runtime.h>
#include <hip/hip_bf16.h>
#include <stdint.h>

// Problem constants (match reference)
#define NUM_PAGES     512
#define PAGE_SIZE     256
#define KV_LORA_RANK  512
#define ROPE_DIM      64
#define CACHE_DIM     576   // KV_LORA_RANK + ROPE_DIM

// ---- bulk copy (async global -> LDS -> global, gfx1250 path) ----
#define COPY_BLOCK 256
#define COPY_WAVES (COPY_BLOCK / 32)
#define CH 8                               // b128 async ops in flight per lane-group per phase
#define WAVE_BYTES (CH * 32 * 16)          // 4 KB staged per wave
#define COPY_GRID 1024                     // 8192 waves -> 9 full tiles each for this size

__global__ __launch_bounds__(COPY_BLOCK)
void cache_bulk_copy_async(const float* __restrict__ src,
                           float* __restrict__ dst,
                           unsigned nchunks /* number of 16B chunks */) {
    __shared__ __align__(16) unsigned char smem[COPY_WAVES * WAVE_BYTES];

    const unsigned lane   = threadIdx.x & 31u;
    const unsigned wave   = threadIdx.x >> 5;
    const unsigned gwave  = blockIdx.x * COPY_WAVES + wave;
    const unsigned nwaves = gridDim.x * COPY_WAVES;

    // Flat LDS pointers on AMDGPU encode the LDS byte offset in the low 32 bits
    const unsigned lds_base = (unsigned)(size_t)(&smem[wave * WAVE_BYTES]);

    const unsigned tile  = CH * 32u;               // chunks consumed per wave-iteration
    const unsigned nfull = nchunks & ~(tile - 1u); // full-tile region

    for (unsigned base = gwave * tile; base < nfull; base += nwaves * tile) {
        // Phase 1: issue CH async B128 loads (global -> LDS), no VGPR data path.
#pragma unroll
        for (int j = 0; j < CH; ++j) {
            unsigned c    = base + (unsigned)(j * 32) + lane;
            unsigned voff = c * 16u;                                  // byte offset < 2^31
            unsigned lds  = lds_base + ((unsigned)(j * 32) + lane) * 16u;
            asm volatile("global_load_async_to_lds_b128 %0, %1, %2 th:TH_LOAD_NT"
                         :: "v"(lds), "v"(voff), "s"(src) : "memory");
        }
        asm volatile("s_wait_asynccnt 0" ::: "memory");   // LDS data landed

        // Phase 2: issue CH async B128 stores (LDS -> global).
#pragma unroll
        for (int j = 0; j < CH; ++j) {
            unsigned c    = base + (unsigned)(j * 32) + lane;
            unsigned voff = c * 16u;
            unsigned lds  = lds_base + ((unsigned)(j * 32) + lane) * 16u;
            asm volatile("global_store_async_from_lds_b128 %0, %1, %2 th:TH_STORE_NT"
                         :: "v"(voff), "v"(lds), "s"(dst) : "memory");
        }
        asm volatile("s_wait_asynccnt 0" ::: "memory");   // LDS reusable next iter
    }

    // Safety tail (not reached for this problem size: nchunks % 256 == 0).
    const unsigned gtid    = blockIdx.x * blockDim.x + threadIdx.x;
    const unsigned gthreads = gridDim.x * blockDim.x;
    for (unsigned c = nfull + gtid; c < nchunks; c += gthreads) {
        ((float4*)dst)[c] = ((const float4*)src)[c];
    }
}

// ---- paged scatter: one block per token, thread i writes float4 i of the slot ----
__global__ __launch_bounds__(160)
void cache_scatter(const float* __restrict__ key,
                   const float* __restrict__ val,
                   const int* __restrict__ pidx,
                   const int* __restrict__ poff,
                   float* __restrict__ out) {
    const unsigned t = blockIdx.x;
    const unsigned i = threadIdx.x;

    // Uniform per block -> scalar loads.
    const long long slot = (long long)pidx[t] * PAGE_SIZE + (long long)poff[t];
    float4* __restrict__ dst = (float4*)(out + slot * (long long)CACHE_DIM); // 16B aligned (slot*2304B)

    if (i < (KV_LORA_RANK / 4)) {                    // 128 float4 from key
        const float4* k4 = (const float4*)(key + (size_t)t * KV_LORA_RANK);
        dst[i] = k4[i];
    } else if (i < (CACHE_DIM / 4)) {                // 16 float4 from value
        const float4* v4 = (const float4*)(val + (size_t)t * ROPE_DIM);
        dst[i] = v4[i - (KV_LORA_RANK / 4)];
    }
}

extern "C" void kernel_launch(void* const* d_in, const int* in_sizes, int n_in,
                              void* d_out, int out_size, void* d_ws, size_t ws_size,
                              hipStream_t stream) {
    // setup_inputs order: k_cache, key_states, value_states, page_idx, page_offset, kv_lora_rank
    const float* k_cache = (const float*)d_in[0];
    const float* key     = (const float*)d_in[1];
    const float* val     = (const float*)d_in[2];
    const int*   pidx    = (const int*)d_in[3];
    const int*   poff    = (const int*)d_in[4];
    float*       out     = (float*)d_out;

    const unsigned nchunks = (unsigned)((size_t)out_size / 4); // float -> float4 count
    const int      tokens  = in_sizes[3];                      // 16384

    // 1) copy entire cache to out (302 MB each way, async-LDS streaming path)
    cache_bulk_copy_async<<<COPY_GRID, COPY_BLOCK, 0, stream>>>(k_cache, out, nchunks);
    // 2) scatter token slots on top (stream order guarantees visibility)
    cache_scatter<<<tokens, 160, 0, stream>>>(key, val, pidx, poff, out);
}